// GCN_13400297963541
// MI455X (gfx1250) — compile-verified
//
#include <hip/hip_runtime.h>

typedef __attribute__((ext_vector_type(2))) float v2f;
typedef __attribute__((ext_vector_type(8))) float v8f;

// Problem constants (match reference setup_inputs)
#define BB 2
#define NN 20000
#define EE 640000
#define HH 128
#define OUTD 10

// ---------------------------------------------------------------------------
// utility: grid-stride float4 zero (all buffer sizes are multiples of 4)
__global__ void zero_f32(float* __restrict__ p, long n4 /* count of float4 */) {
    float4* p4 = (float4*)p;
    long i = (long)blockIdx.x * blockDim.x + threadIdx.x;
    long stride = (long)gridDim.x * blockDim.x;
    float4 z = {0.f, 0.f, 0.f, 0.f};
    for (; i < n4; i += stride) p4[i] = z;
}

// degree count: atomicAdd 1.0 per edge destination
__global__ void deg_kernel(const long long* __restrict__ ei, float* __restrict__ deg, int E) {
    int e = blockIdx.x * blockDim.x + threadIdx.x;
    if (e < E) atomicAdd(&deg[(int)ei[E + e]], 1.0f);
}

// counts -> dinv, with the +1 self-loop folded in (deg >= 1 always)
__global__ void dinv_kernel(float* __restrict__ d, int N) {
    int i = blockIdx.x * blockDim.x + threadIdx.x;
    if (i < N) d[i] = rsqrtf(d[i] + 1.0f);
}

// per-edge symmetric norm
__global__ void norm_kernel(const long long* __restrict__ ei, const float* __restrict__ dinv,
                            float* __restrict__ norm, int E) {
    int e = blockIdx.x * blockDim.x + threadIdx.x;
    if (e < E) norm[e] = dinv[(int)ei[e]] * dinv[(int)ei[E + e]];
}

// ---------------------------------------------------------------------------
// C[M x 128] = A[M x 128] @ W[128 x 128], fp32 WMMA 16x16x4.
// One block = 256 threads = 8 waves; block computes 16 rows x 128 cols;
// wave w computes the 16x16 tile at columns [16w, 16w+16).
__global__ __launch_bounds__(256) void gemm128_wmma(const float* __restrict__ A,
                                                    const float* __restrict__ W,
                                                    float* __restrict__ C, int M) {
    __shared__ float As[16][132];  // stride 132 -> bank-conflict-free column reads
    const int m0  = blockIdx.x * 16;
    const int tid = threadIdx.x;
    for (int i = tid; i < 16 * 128; i += 256) {
        int r = i >> 7, c = i & 127;
        int gr = m0 + r;
        As[r][c] = (gr < M) ? A[(long)gr * 128 + c] : 0.0f;
    }
    __syncthreads();

    const int wave  = tid >> 5;
    const int lane  = tid & 31;
    const int n0    = wave * 16;
    const int mrow  = lane & 15;
    const int khalf = (lane >> 4) << 1;  // 0 for lanes 0-15, 2 for lanes 16-31

    v8f acc = {};
    for (int k = 0; k < 128; k += 4) {
        v2f a, b;
        a.x = As[mrow][k + khalf];
        a.y = As[mrow][k + khalf + 1];
        const float* Wp = W + (long)(k + khalf) * 128 + n0 + mrow;
        b.x = Wp[0];
        b.y = Wp[128];
        acc = __builtin_amdgcn_wmma_f32_16x16x4_f32(false, a, false, b,
                                                    (short)0, acc, false, false);
    }
    const int mbase = m0 + ((lane >> 4) << 3);
    const int nn    = n0 + mrow;
#pragma unroll
    for (int v = 0; v < 8; ++v) {
        int gm = mbase + v;
        if (gm < M) C[(long)gm * 128 + nn] = acc[v];
    }
}

// ---------------------------------------------------------------------------
// Edge aggregation: one wave per edge; each lane handles 4 channels (float4).
__global__ __launch_bounds__(256) void agg_kernel(const long long* __restrict__ ei,
                                                  const float* __restrict__ norm,
                                                  const float* __restrict__ h,
                                                  float* __restrict__ agg, int E) {
    long gid  = (long)blockIdx.x * blockDim.x + threadIdx.x;
    int  e    = (int)(gid >> 5);
    int  lane = threadIdx.x & 31;
    if (e >= E) return;
    int   src = (int)ei[e];
    int   dst = (int)ei[E + e];
    float w   = norm[e];
    // prefetch the batch-1 source row while working on batch 0
    __builtin_prefetch(h + ((long)NN + src) * HH + lane * 4, 0, 0);
#pragma unroll
    for (int b = 0; b < BB; ++b) {
        const float4 g = *((const float4*)(h + ((long)b * NN + src) * HH) + lane);
        float* ap = agg + ((long)b * NN + dst) * HH + lane * 4;
        atomicAdd(ap + 0, g.x * w);
        atomicAdd(ap + 1, g.y * w);
        atomicAdd(ap + 2, g.z * w);
        atomicAdd(ap + 3, g.w * w);
    }
}

// self-loop contribution + bias + ReLU, fused (in place on agg)
__global__ void sbr_kernel(const float* __restrict__ hlin, const float* __restrict__ dinv,
                           const float* __restrict__ bias, float* __restrict__ agg,
                           long total) {
    long idx = (long)blockIdx.x * blockDim.x + threadIdx.x;
    if (idx >= total) return;
    int   c    = (int)(idx & (HH - 1));
    int   node = (int)((idx >> 7) % NN);
    float di   = dinv[node];
    float v    = agg[idx] + di * di * hlin[idx] + bias[c];
    agg[idx]   = fmaxf(v, 0.0f);
}

// ---------------------------------------------------------------------------
// BatchNorm stats: per-channel sum / sumsq over rows (register accumulate,
// then two atomics per thread). blockDim must be 128 (one thread per channel).
__global__ __launch_bounds__(128) void bn_stats(const float* __restrict__ h,
                                                float* __restrict__ stats, long rows) {
    int   c  = threadIdx.x;
    float s  = 0.f, s2 = 0.f;
    for (long r = blockIdx.x; r < rows; r += gridDim.x) {
        float v = h[r * HH + c];
        s  += v;
        s2 += v * v;
    }
    atomicAdd(&stats[c], s);
    atomicAdd(&stats[HH + c], s2);
}

__global__ void bn_apply(float* __restrict__ h, const float* __restrict__ stats,
                         const float* __restrict__ gamma, const float* __restrict__ beta,
                         long total, float invM) {
    long idx = (long)blockIdx.x * blockDim.x + threadIdx.x;
    if (idx >= total) return;
    int   c    = (int)(idx & (HH - 1));
    float mean = stats[c] * invM;
    float var  = stats[HH + c] * invM - mean * mean;  // biased, matches jnp.var
    h[idx] = (h[idx] - mean) * rsqrtf(var + 1e-5f) * gamma[c] + beta[c];
}

// ---------------------------------------------------------------------------
// Classifier: out[M x 10] = A[M x 128] @ Wc[128 x 10] + bc, WMMA with Wc
// zero-padded to 16 columns in LDS. Block = 256 thr = 8 waves = 128 rows.
__global__ __launch_bounds__(256) void classifier_wmma(const float* __restrict__ A,
                                                       const float* __restrict__ Wc,
                                                       const float* __restrict__ bc,
                                                       float* __restrict__ out, int M) {
    __shared__ float Ws[128][16];
    for (int i = threadIdx.x; i < 128 * 16; i += 256) {
        int r = i >> 4, c = i & 15;
        Ws[r][c] = (c < OUTD) ? Wc[r * OUTD + c] : 0.0f;
    }
    __syncthreads();

    const int wave = threadIdx.x >> 5;
    const int lane = threadIdx.x & 31;
    const int m0   = blockIdx.x * 128 + wave * 16;
    if (m0 >= M) return;  // whole-wave exit: EXEC stays all-1 for WMMA
    const int mrow  = lane & 15;
    const int khalf = (lane >> 4) << 1;

    v8f acc = {};
    for (int k = 0; k < 128; k += 4) {
        v2f a, b;
        const float* Ap = A + (long)(m0 + mrow) * 128 + k + khalf;
        a.x = Ap[0];
        a.y = Ap[1];
        b.x = Ws[k + khalf][mrow];
        b.y = Ws[k + khalf + 1][mrow];
        acc = __builtin_amdgcn_wmma_f32_16x16x4_f32(false, a, false, b,
                                                    (short)0, acc, false, false);
    }
    const int mbase = m0 + ((lane >> 4) << 3);
    const int nn    = mrow;
    if (nn < OUTD) {
        float bias = bc[nn];
#pragma unroll
        for (int v = 0; v < 8; ++v) {
            int gm = mbase + v;
            if (gm < M) out[(long)gm * OUTD + nn] = acc[v] + bias;
        }
    }
}

// ---------------------------------------------------------------------------
extern "C" void kernel_launch(void* const* d_in, const int* in_sizes, int n_in,
                              void* d_out, int out_size, void* d_ws, size_t ws_size,
                              hipStream_t stream) {
    const float*     x   = (const float*)d_in[0];
    const float*     W1  = (const float*)d_in[1];
    const float*     b1  = (const float*)d_in[2];
    const float*     W2  = (const float*)d_in[3];
    const float*     b2  = (const float*)d_in[4];
    const float*     g1  = (const float*)d_in[5];
    const float*     bt1 = (const float*)d_in[6];
    const float*     g2  = (const float*)d_in[7];
    const float*     bt2 = (const float*)d_in[8];
    const float*     Wc  = (const float*)d_in[9];
    const float*     bc  = (const float*)d_in[10];
    const long long* ei  = (const long long*)d_in[11];  // int64 per reference

    const int  M     = BB * NN;           // 40000 rows
    const long total = (long)M * HH;      // 5,120,000 elems

    // workspace carve (floats): dinv | norm | bufA | bufB | stats
    float* ws    = (float*)d_ws;
    float* dinv  = ws;                    // NN
    float* norm  = dinv + NN;             // EE
    float* bufA  = norm + EE;             // M*HH
    float* bufB  = bufA + total;          // M*HH
    float* stats = bufB + total;          // 256

    const dim3 T256(256), T128(128);
    const int  aggBlocks  = (EE * 32 + 255) / 256;
    const int  elemBlocks = (int)((total + 255) / 256);

    // ---- graph normalization ----
    zero_f32<<<64, T256, 0, stream>>>(dinv, NN / 4);
    deg_kernel<<<(EE + 255) / 256, T256, 0, stream>>>(ei, dinv, EE);
    dinv_kernel<<<(NN + 255) / 256, T256, 0, stream>>>(dinv, NN);
    norm_kernel<<<(EE + 255) / 256, T256, 0, stream>>>(ei, dinv, norm, EE);

    // ---- layer 1: conv -> relu -> bn ----
    gemm128_wmma<<<M / 16, T256, 0, stream>>>(x, W1, bufA, M);
    zero_f32<<<2048, T256, 0, stream>>>(bufB, total / 4);
    agg_kernel<<<aggBlocks, T256, 0, stream>>>(ei, norm, bufA, bufB, EE);
    sbr_kernel<<<elemBlocks, T256, 0, stream>>>(bufA, dinv, b1, bufB, total);
    zero_f32<<<1, T256, 0, stream>>>(stats, 256 / 4);
    bn_stats<<<512, T128, 0, stream>>>(bufB, stats, M);
    bn_apply<<<elemBlocks, T256, 0, stream>>>(bufB, stats, g1, bt1, total, 1.0f / (float)M);

    // ---- layer 2: conv -> relu -> bn ----
    gemm128_wmma<<<M / 16, T256, 0, stream>>>(bufB, W2, bufA, M);
    zero_f32<<<2048, T256, 0, stream>>>(bufB, total / 4);
    agg_kernel<<<aggBlocks, T256, 0, stream>>>(ei, norm, bufA, bufB, EE);
    sbr_kernel<<<elemBlocks, T256, 0, stream>>>(bufA, dinv, b2, bufB, total);
    zero_f32<<<1, T256, 0, stream>>>(stats, 256 / 4);
    bn_stats<<<512, T128, 0, stream>>>(bufB, stats, M);
    bn_apply<<<elemBlocks, T256, 0, stream>>>(bufB, stats, g2, bt2, total, 1.0f / (float)M);

    // ---- classifier ----
    classifier_wmma<<<(M + 127) / 128, T256, 0, stream>>>(bufB, Wc, bc, (float*)d_out, M);
}